// InteractionBlock_72146860638426
// MI455X (gfx1250) — compile-verified
//
#include <hip/hip_runtime.h>
#include <hip/hip_bf16.h>

typedef __attribute__((ext_vector_type(2))) float v2f;
typedef __attribute__((ext_vector_type(8))) float v8f;

#define WMMA_F32(a, b, c) \
  __builtin_amdgcn_wmma_f32_16x16x4_f32(false, (a), false, (b), (short)0, (c), false, false)

#define NN 50000
#define EE 800000

// Weight repack: element (k, col) of a (K x 32) matrix stored at
//   (k>>1)*64 + col*2 + (k&1)
// so a B fragment {(k,col),(k+1,col)} with even k is one contiguous v2f.
__device__ __forceinline__ int wpack(int k, int col) {
  return ((k >> 1) << 6) + (col << 1) + (k & 1);
}

// ---------------------------------------------------------------------------
// Kernel: zero the scatter accumulators (n_s, n_v) in workspace.
// ---------------------------------------------------------------------------
__global__ void zero_accum(float* __restrict__ p, size_t n) {
  size_t i = (size_t)blockIdx.x * blockDim.x + threadIdx.x;
  size_t stride = (size_t)gridDim.x * blockDim.x;
  for (; i < n; i += stride) p[i] = 0.0f;
}

// ---------------------------------------------------------------------------
// Kernel: per-node pre-pass.  Per 16-node tile (one wave each):
//   s1  = s @ W1_s * (1/sqrt(32))                  (16x32 @ 32x32)
//   v1  = v_i @ W1_v * (1/sqrt(32))   i=0..2
//   sc_s= (s (x) attrs) @ Wsc_s * (1/sqrt(128))    (16x128 @ 128x32)
//   sc_v= (v_i (x) attrs) @ Wsc_v * (1/sqrt(128))
// V_WMMA_F32_16X16X4_F32; block = 64 threads = 2 waves = 2 tiles.
// ---------------------------------------------------------------------------
__global__ void __launch_bounds__(64, 1)
node_pre(const float* __restrict__ nf,     // (N,128)
         const float* __restrict__ nattr,  // (N,4)
         const float* __restrict__ W1s,    // (32,32)
         const float* __restrict__ W1v,    // (32,32)
         const float* __restrict__ Wscs,   // (32,4,32) -> (128,32)
         const float* __restrict__ Wscv,   // (128,32)
         float* __restrict__ s1,           // (N,32)
         float* __restrict__ v1,           // (N,32,3) as n*96 + u*3 + i
         float* __restrict__ scs,          // (N,32)
         float* __restrict__ scv,          // (N,32,3)
         int Nn) {
  __shared__ __align__(16) float lW1s[1024], lW1v[1024], lWscs[4096], lWscv[4096];
  __shared__ __align__(16) float lS[2][16 * 32];
  __shared__ __align__(16) float lV[2][16 * 96];
  __shared__ __align__(16) float lA[2][16 * 4];

  const int t = threadIdx.x;
  for (int i = t; i < 1024; i += 64) {
    int k = i >> 5, c = i & 31, d = wpack(k, c);
    lW1s[d] = W1s[i];
    lW1v[d] = W1v[i];
  }
  for (int i = t; i < 4096; i += 64) {
    int k = i >> 5, c = i & 31, d = wpack(k, c);
    lWscs[d] = Wscs[i];
    lWscv[d] = Wscv[i];
  }
  __syncthreads();  // all threads reach this before any divergence

  const int wave = t >> 5, lane = t & 31;
  const int tile = blockIdx.x * 2 + wave;
  if (tile * 16 >= Nn) return;  // wave-uniform
  const int m0 = tile * 16;

  float* S = lS[wave];
  float* V = lV[wave];
  float* A = lA[wave];
  // stage this wave's 16-node tile (coalesced); same-wave LDS is in-order
  for (int i = lane; i < 512; i += 32) {
    int r = i >> 5, c = i & 31;
    S[i] = nf[(size_t)(m0 + r) * 128 + c];
  }
  for (int i = lane; i < 1536; i += 32) {
    int r = i / 96, c = i % 96;
    V[i] = nf[(size_t)(m0 + r) * 128 + 32 + c];
  }
  for (int i = lane; i < 64; i += 32) {
    int r = i >> 2, c = i & 3;
    A[i] = nattr[(size_t)(m0 + r) * 4 + c];
  }

  const int hf = lane >> 4;   // wave half
  const int xr = lane & 15;   // A row / B,C column
  const int koff = 2 * hf;    // K offset of this half within a k-step
  const float lin = 0.17677669529663687f;  // 1/sqrt(32)
  const float inv = 0.08838834764831845f;  // 1/sqrt(128)

  // pair-packed weight views: pair p, col c at [p*32 + c]
  const v2f* W1sp  = (const v2f*)lW1s;
  const v2f* W1vp  = (const v2f*)lW1v;
  const v2f* Wscsp = (const v2f*)lWscs;
  const v2f* Wscvp = (const v2f*)lWscv;

  // ---- s1 = S @ W1_s ----
  {
    v8f acc0 = {}, acc1 = {};
    for (int k0 = 0; k0 < 32; k0 += 4) {
      int p = (k0 + koff) >> 1;
      v2f a = *(const v2f*)&S[xr * 32 + k0 + koff];
      v2f b0 = W1sp[p * 32 + xr];
      v2f b1 = W1sp[p * 32 + 16 + xr];
      acc0 = WMMA_F32(a, b0, acc0);
      acc1 = WMMA_F32(a, b1, acc1);
    }
    for (int r = 0; r < 8; ++r) {
      int row = m0 + r + 8 * hf;
      s1[(size_t)row * 32 + xr]      = acc0[r] * lin;
      s1[(size_t)row * 32 + 16 + xr] = acc1[r] * lin;
    }
  }

  // ---- v1_i = V_i @ W1_v ----
  for (int i = 0; i < 3; ++i) {
    v8f acc0 = {}, acc1 = {};
    for (int k0 = 0; k0 < 32; k0 += 4) {
      int p = (k0 + koff) >> 1;
      v2f a;
      a.x = V[xr * 96 + (k0 + koff) * 3 + i];
      a.y = V[xr * 96 + (k0 + koff + 1) * 3 + i];
      v2f b0 = W1vp[p * 32 + xr];
      v2f b1 = W1vp[p * 32 + 16 + xr];
      acc0 = WMMA_F32(a, b0, acc0);
      acc1 = WMMA_F32(a, b1, acc1);
    }
    for (int r = 0; r < 8; ++r) {
      int row = m0 + r + 8 * hf;
      v1[(size_t)row * 96 + xr * 3 + i]        = acc0[r] * lin;
      v1[(size_t)row * 96 + (16 + xr) * 3 + i] = acc1[r] * lin;
    }
  }

  // ---- sc_s = (s ⊗ attrs) @ Wsc_s,  K = 128, A built on the fly ----
  {
    v8f acc0 = {}, acc1 = {};
    for (int k0 = 0; k0 < 128; k0 += 4) {
      int kk0 = k0 + koff, kk1 = kk0 + 1;
      int p = kk0 >> 1;
      v2f a;
      a.x = S[xr * 32 + (kk0 >> 2)] * A[xr * 4 + (kk0 & 3)];
      a.y = S[xr * 32 + (kk1 >> 2)] * A[xr * 4 + (kk1 & 3)];
      v2f b0 = Wscsp[p * 32 + xr];
      v2f b1 = Wscsp[p * 32 + 16 + xr];
      acc0 = WMMA_F32(a, b0, acc0);
      acc1 = WMMA_F32(a, b1, acc1);
    }
    for (int r = 0; r < 8; ++r) {
      int row = m0 + r + 8 * hf;
      scs[(size_t)row * 32 + xr]      = acc0[r] * inv;
      scs[(size_t)row * 32 + 16 + xr] = acc1[r] * inv;
    }
  }

  // ---- sc_v_i = (v_i ⊗ attrs) @ Wsc_v ----
  for (int i = 0; i < 3; ++i) {
    v8f acc0 = {}, acc1 = {};
    for (int k0 = 0; k0 < 128; k0 += 4) {
      int kk0 = k0 + koff, kk1 = kk0 + 1;
      int p = kk0 >> 1;
      v2f a;
      a.x = V[xr * 96 + (kk0 >> 2) * 3 + i] * A[xr * 4 + (kk0 & 3)];
      a.y = V[xr * 96 + (kk1 >> 2) * 3 + i] * A[xr * 4 + (kk1 & 3)];
      v2f b0 = Wscvp[p * 32 + xr];
      v2f b1 = Wscvp[p * 32 + 16 + xr];
      acc0 = WMMA_F32(a, b0, acc0);
      acc1 = WMMA_F32(a, b1, acc1);
    }
    for (int r = 0; r < 8; ++r) {
      int row = m0 + r + 8 * hf;
      scv[(size_t)row * 96 + xr * 3 + i]        = acc0[r] * inv;
      scv[(size_t)row * 96 + (16 + xr) * 3 + i] = acc1[r] * inv;
    }
  }
}

// ---------------------------------------------------------------------------
// Kernel: per-edge message pass.  One wave per edge; lane == channel u.
// Fuses the 8->8->128 MLP, the neighbor gather, the elementwise message,
// and the scatter-add (global_atomic_add_f32, resolves in L2).
// ---------------------------------------------------------------------------
__global__ void edge_kernel(const float* __restrict__ ee,       // (E,8)
                            const float* __restrict__ eattr,    // (E,4)
                            const long long* __restrict__ eidx, // (2,E) int64
                            const float* __restrict__ Wfc1,     // (8,8)
                            const float* __restrict__ Wfc2,     // (8,128)
                            const float* __restrict__ s1,       // (N,32)
                            const float* __restrict__ v1,       // (N,32,3)
                            float* __restrict__ ns,             // (N,64) accum
                            float* __restrict__ nv,             // (N,64,3) accum
                            int E_) {
  __shared__ float lWfc1[64], lWfc2[1024];
  for (int i = threadIdx.x; i < 64; i += blockDim.x) lWfc1[i] = Wfc1[i];
  for (int i = threadIdx.x; i < 1024; i += blockDim.x) lWfc2[i] = Wfc2[i];
  __syncthreads();

  const int lane = threadIdx.x & 31;
  const long long e = (long long)blockIdx.x * (blockDim.x >> 5) + (threadIdx.x >> 5);
  if (e >= E_) return;  // wave-uniform

  const long long ctr = eidx[e];
  const long long nbr = eidx[(long long)E_ + e];
  // warm the gather rows (global_prefetch_b8) while the MLP computes
  __builtin_prefetch(&s1[nbr * 32 + lane], 0, 3);
  __builtin_prefetch(&v1[nbr * 96 + lane * 3], 0, 3);

  // Bessel row: lanes 0..7 load, everyone reads via shuffle
  float my = (lane < 8) ? ee[e * 8 + lane] : 0.0f;
  const float c1 = 0.3535533905932738f;  // 1/sqrt(8)
  float h[8];
  for (int k = 0; k < 8; ++k) {
    float acc = 0.0f;
    for (int j = 0; j < 8; ++j) acc += __shfl(my, j, 32) * lWfc1[j * 8 + k];
    acc *= c1;
    h[k] = acc / (1.0f + __expf(-acc));  // SiLU
  }
  float w0 = 0.f, w1 = 0.f, w2 = 0.f, w3 = 0.f;
  for (int k = 0; k < 8; ++k) {
    float hk = h[k];
    w0 += hk * lWfc2[k * 128 + lane];
    w1 += hk * lWfc2[k * 128 + 32 + lane];
    w2 += hk * lWfc2[k * 128 + 64 + lane];
    w3 += hk * lWfc2[k * 128 + 96 + lane];
  }
  w0 *= c1; w1 *= c1; w2 *= c1; w3 *= c1;  // 1/sqrt(HID) == 1/sqrt(8)

  const float xs  = s1[nbr * 32 + lane];
  const float xv0 = v1[nbr * 96 + lane * 3 + 0];
  const float xv1 = v1[nbr * 96 + lane * 3 + 1];
  const float xv2 = v1[nbr * 96 + lane * 3 + 2];
  const float es  = eattr[e * 4 + 0];
  const float ev0 = eattr[e * 4 + 1];
  const float ev1 = eattr[e * 4 + 2];
  const float ev2 = eattr[e * 4 + 3];

  const float inv3 = 0.5773502691896258f;  // 1/sqrt(3)
  const float os0 = w0 * xs * es;
  const float dot = xv0 * ev0 + xv1 * ev1 + xv2 * ev2;
  const float os3 = w3 * dot * inv3;
  const float a1 = w1 * xs;
  const float a2 = w2 * es;

  float* nsrow = ns + ctr * 64;
  atomicAdd(nsrow + lane, os0);
  atomicAdd(nsrow + 32 + lane, os3);
  float* nvrow = nv + ctr * 192;
  atomicAdd(nvrow + lane * 3 + 0, a1 * ev0);
  atomicAdd(nvrow + lane * 3 + 1, a1 * ev1);
  atomicAdd(nvrow + lane * 3 + 2, a1 * ev2);
  atomicAdd(nvrow + 96 + lane * 3 + 0, a2 * xv0);
  atomicAdd(nvrow + 96 + lane * 3 + 1, a2 * xv1);
  atomicAdd(nvrow + 96 + lane * 3 + 2, a2 * xv2);
}

// ---------------------------------------------------------------------------
// Kernel: per-node post-pass.  out_s = n_s @ W2_s * lin2 + sc_s (K=64)
//                              out_v = n_v_i @ W2_v * lin2 + sc_v (K=64)
// Writes the final (N,128) output.  WMMA f32 16x16x4, one wave per 16 nodes.
// ---------------------------------------------------------------------------
__global__ void __launch_bounds__(64, 1)
node_post(const float* __restrict__ ns,   // (N,64)
          const float* __restrict__ nv,   // (N,64,3)
          const float* __restrict__ scs,  // (N,32)
          const float* __restrict__ scv,  // (N,32,3)
          const float* __restrict__ W2s,  // (64,32)
          const float* __restrict__ W2v,  // (64,32)
          float* __restrict__ out,        // (N,128)
          int Nn) {
  __shared__ __align__(16) float lW2s[2048], lW2v[2048];
  __shared__ __align__(16) float lNS[2][16 * 64];
  __shared__ __align__(16) float lNV[2][16 * 192];

  const int t = threadIdx.x;
  for (int i = t; i < 2048; i += 64) {
    int k = i >> 5, c = i & 31, d = wpack(k, c);
    lW2s[d] = W2s[i];
    lW2v[d] = W2v[i];
  }
  __syncthreads();

  const int wave = t >> 5, lane = t & 31;
  const int tile = blockIdx.x * 2 + wave;
  if (tile * 16 >= Nn) return;  // wave-uniform
  const int m0 = tile * 16;

  float* NS = lNS[wave];
  float* NV = lNV[wave];
  for (int i = lane; i < 1024; i += 32) {
    int r = i >> 6, c = i & 63;
    NS[i] = ns[(size_t)(m0 + r) * 64 + c];
  }
  for (int i = lane; i < 3072; i += 32) {
    int r = i / 192, c = i % 192;
    NV[i] = nv[(size_t)(m0 + r) * 192 + c];
  }

  const int hf = lane >> 4;
  const int xr = lane & 15;
  const int koff = 2 * hf;
  const float lin2 = 0.125f;  // 1/sqrt(64)

  const v2f* W2sp = (const v2f*)lW2s;
  const v2f* W2vp = (const v2f*)lW2v;

  // ---- out_s ----
  {
    v8f acc0 = {}, acc1 = {};
    for (int k0 = 0; k0 < 64; k0 += 4) {
      int p = (k0 + koff) >> 1;
      v2f a = *(const v2f*)&NS[xr * 64 + k0 + koff];
      v2f b0 = W2sp[p * 32 + xr];
      v2f b1 = W2sp[p * 32 + 16 + xr];
      acc0 = WMMA_F32(a, b0, acc0);
      acc1 = WMMA_F32(a, b1, acc1);
    }
    for (int r = 0; r < 8; ++r) {
      int row = m0 + r + 8 * hf;
      out[(size_t)row * 128 + xr] =
          acc0[r] * lin2 + scs[(size_t)row * 32 + xr];
      out[(size_t)row * 128 + 16 + xr] =
          acc1[r] * lin2 + scs[(size_t)row * 32 + 16 + xr];
    }
  }

  // ---- out_v (3 components) ----
  for (int i = 0; i < 3; ++i) {
    v8f acc0 = {}, acc1 = {};
    for (int k0 = 0; k0 < 64; k0 += 4) {
      int p = (k0 + koff) >> 1;
      v2f a;
      a.x = NV[xr * 192 + (k0 + koff) * 3 + i];
      a.y = NV[xr * 192 + (k0 + koff + 1) * 3 + i];
      v2f b0 = W2vp[p * 32 + xr];
      v2f b1 = W2vp[p * 32 + 16 + xr];
      acc0 = WMMA_F32(a, b0, acc0);
      acc1 = WMMA_F32(a, b1, acc1);
    }
    for (int r = 0; r < 8; ++r) {
      int row = m0 + r + 8 * hf;
      out[(size_t)row * 128 + 32 + xr * 3 + i] =
          acc0[r] * lin2 + scv[(size_t)row * 96 + xr * 3 + i];
      out[(size_t)row * 128 + 32 + (16 + xr) * 3 + i] =
          acc1[r] * lin2 + scv[(size_t)row * 96 + (16 + xr) * 3 + i];
    }
  }
}

// ---------------------------------------------------------------------------
extern "C" void kernel_launch(void* const* d_in, const int* in_sizes, int n_in,
                              void* d_out, int out_size, void* d_ws, size_t ws_size,
                              hipStream_t stream) {
  const float* edge_embedding = (const float*)d_in[0];
  const float* node_attrs     = (const float*)d_in[1];
  const float* node_features  = (const float*)d_in[2];
  const long long* edge_index = (const long long*)d_in[3];  // int64
  const float* edge_attrs     = (const float*)d_in[4];
  const float* W1s  = (const float*)d_in[5];
  const float* W1v  = (const float*)d_in[6];
  const float* Wfc1 = (const float*)d_in[7];
  const float* Wfc2 = (const float*)d_in[8];
  const float* W2s  = (const float*)d_in[9];
  const float* W2v  = (const float*)d_in[10];
  const float* Wscs = (const float*)d_in[11];
  const float* Wscv = (const float*)d_in[12];
  float* out = (float*)d_out;

  const int Nn = NN, Ee = EE;

  // workspace layout (floats): s1 | v1 | sc_s | sc_v | n_s | n_v
  float* ws  = (float*)d_ws;
  float* s1  = ws;                     // N*32
  float* v1  = ws + (size_t)Nn * 32;   // N*96
  float* scs = ws + (size_t)Nn * 128;  // N*32
  float* scv = ws + (size_t)Nn * 160;  // N*96
  float* ns  = ws + (size_t)Nn * 256;  // N*64
  float* nv  = ws + (size_t)Nn * 320;  // N*192

  // 1) zero scatter accumulators (n_s + n_v = N*256 floats)
  zero_accum<<<2048, 256, 0, stream>>>(ns, (size_t)Nn * 256);

  // 2) node pre-pass: 3125 tiles of 16 nodes, 2 waves/block
  const int tiles = (Nn + 15) / 16;
  node_pre<<<(tiles + 1) / 2, 64, 0, stream>>>(node_features, node_attrs,
                                               W1s, W1v, Wscs, Wscv,
                                               s1, v1, scs, scv, Nn);

  // 3) edge pass: one wave per edge, 8 edges per 256-thread block
  edge_kernel<<<(Ee + 7) / 8, 256, 0, stream>>>(edge_embedding, edge_attrs,
                                                edge_index, Wfc1, Wfc2,
                                                s1, v1, ns, nv, Ee);

  // 4) node post-pass
  node_post<<<(tiles + 1) / 2, 64, 0, stream>>>(ns, nv, scs, scv,
                                                W2s, W2v, out, Nn);
}